// AlgorithmicRhythmTeacher_16784732193416
// MI455X (gfx1250) — compile-verified
//
#include <hip/hip_runtime.h>
#include <stdint.h>

#define NT 256
#define TMAX 8192

typedef __attribute__((ext_vector_type(2))) float v2f;
typedef __attribute__((ext_vector_type(8))) float v8f;

// ---------------------------------------------------------------------------
// Pre-kernel: ratio = clip(max(0.3, max_b clip(stats[b,4],0,1)), 0, 1)
// ---------------------------------------------------------------------------
__global__ void ratio_kernel(const float* __restrict__ stats, float* __restrict__ ws, int B) {
    __shared__ float red[8];
    float mx = 0.f;
    for (int i = threadIdx.x; i < B; i += blockDim.x) {
        float v = stats[(size_t)i * 5 + 4];
        v = fminf(fmaxf(v, 0.f), 1.f);
        mx = fmaxf(mx, v);
    }
    for (int o = 16; o > 0; o >>= 1) mx = fmaxf(mx, __shfl_down(mx, o, 32));
    int lane = threadIdx.x & 31, wid = threadIdx.x >> 5;
    if (lane == 0) red[wid] = mx;
    __syncthreads();
    if (threadIdx.x == 0) {
        float m = 0.3f;
        int nw = (blockDim.x + 31) / 32;
        for (int i = 0; i < nw; i++) m = fmaxf(m, red[i]);
        ws[0] = fminf(fmaxf(m, 0.f), 1.f);
    }
}

// ---------------------------------------------------------------------------
// Block-wide sum of 256 per-thread partials; result valid on thread 0 only.
// WMMA path: D = ones(16x4) x B(4x16) + C chained over 4 groups of 64
// partials reduces everything to 16 column sums (layout-invariant for a
// total sum), finished with 4 shuffles. Wave 0 runs with full EXEC as
// required by v_wmma.
// ---------------------------------------------------------------------------
__device__ __forceinline__ float blockSum(float v, float* part) {
    part[threadIdx.x] = v;
    __syncthreads();
    float r = 0.f;
#if __has_builtin(__builtin_amdgcn_wmma_f32_16x16x4_f32)
    if (threadIdx.x < 32) {
        int lane = threadIdx.x;
        v2f ones; ones.x = 1.f; ones.y = 1.f;
        v8f acc = {};
        for (int g = 0; g < 4; g++) {
            v2f bm;
            bm.x = part[g * 64 + lane];
            bm.y = part[g * 64 + 32 + lane];
            acc = __builtin_amdgcn_wmma_f32_16x16x4_f32(
                false, ones, false, bm, (short)0, acc, false, false);
        }
        float cs = acc[0];                    // colsum[lane] for lanes 0..15
        cs += __shfl_down(cs, 8, 32);
        cs += __shfl_down(cs, 4, 32);
        cs += __shfl_down(cs, 2, 32);
        cs += __shfl_down(cs, 1, 32);
        r = cs;                               // lane 0 holds the full sum
    }
#else
    {
        float w = part[threadIdx.x];
        for (int o = 16; o > 0; o >>= 1) w += __shfl_down(w, o, 32);
        __syncthreads();
        if ((threadIdx.x & 31) == 0) part[threadIdx.x >> 5] = w;
        __syncthreads();
        if (threadIdx.x == 0)
            for (int i = 0; i < NT / 32; i++) r += part[i];
    }
#endif
    __syncthreads();
    return r;
}

__device__ __forceinline__ float sigmoidf(float x) { return 1.f / (1.f + expf(-x)); }

// ---------------------------------------------------------------------------
// Main kernel: one workgroup per batch row.
// ---------------------------------------------------------------------------
__global__ __launch_bounds__(NT) void teacher_kernel(
    const float* __restrict__ dur, const float* __restrict__ stats,
    const float* __restrict__ trace, const float* __restrict__ mask,
    const float* __restrict__ cue, float* __restrict__ out,
    const float* __restrict__ ws, int B, int T)
{
    __shared__ float lds_dur[TMAX];
    __shared__ float lds_mask[TMAX];
    __shared__ float s_speech[TMAX];   // raw speech scores (pre-smooth)
    __shared__ float s_pause[TMAX];    // pause scores, later kept-only
    __shared__ float s_tot[NT];        // reduce partials / scan offsets
    __shared__ unsigned int hist[256];
    __shared__ float sc[20];
    __shared__ unsigned int su[2];     // su[0]=threshold bit prefix, su[1]=remaining k

    const int tid = threadIdx.x;
    const int b   = blockIdx.x;
    const int E   = T / NT;            // elements per thread (blocked layout)
    const int base = tid * E;
    const size_t rowf = (size_t)b * T;
    const float* durR  = dur  + rowf;
    const float* maskR = mask + rowf;
    const float* cueR  = cue  + rowf;
    const float* trR   = trace + rowf * 4;

    const size_t BT = (size_t)B * T;
    float* out_se = out;               // speech_exec   [B,T]
    float* out_pe = out + BT;          // pause_exec    [B,T]
    float* out_sb = out + 2 * BT;      // speech_budget [B]
    float* out_pb = out_sb + B;        // pause_budget  [B]
    float* out_al = out_pb + B;        // allocation    [B,T]
    float* out_cf = out_al + BT;       // confidence    [B]
    float* out_tc = out_cf + B;        // trace_ctx     [B,T,4]
    float* out_pc = out_tc + 4 * BT;   // prefix_clock  [B,T]
    float* out_bk = out_pc + BT;       // prefix_backlog[B,T]

    // ------------------------------------------------------------------
    // Stage dur & mask rows into LDS with CDNA5 async global->LDS copies.
    // Each lane moves 16B chunks; tracked by ASYNCcnt.
    // ------------------------------------------------------------------
    {
        uint32_t doff = (uint32_t)(uintptr_t)&lds_dur[0];
        uint32_t moff = (uint32_t)(uintptr_t)&lds_mask[0];
        int nchunk = T / 4;            // 16-byte chunks
        for (int c = tid; c < nchunk; c += NT) {
            uint32_t la = doff + (uint32_t)c * 16u;
            const float* ga = durR + c * 4;
            asm volatile("global_load_async_to_lds_b128 %0, %1, off"
                         :: "v"(la), "v"(ga) : "memory");
            uint32_t lb = moff + (uint32_t)c * 16u;
            const float* gb = maskR + c * 4;
            asm volatile("global_load_async_to_lds_b128 %0, %1, off"
                         :: "v"(lb), "v"(gb) : "memory");
        }
        asm volatile("s_wait_asynccnt 0" ::: "memory");
    }
    __syncthreads();

    // ------------------------------------------------------------------
    // Pass 1: masked sums / sums of squares.
    // ------------------------------------------------------------------
    float a_sm=0,a_sd=0,a_s0=0,a_s1=0,a_s2=0,a_s3=0,a_q0=0,a_q1=0,a_q2=0,a_q3=0,a_sc=0,a_qc=0;
    for (int j = 0; j < E; j++) {
        int t = base + j;
        float m = lds_mask[t];
        float d = lds_dur[t];
        float4 tr = ((const float4*)trR)[t];
        float c = cueR[t];
        a_sm += m;          a_sd += d * m;
        a_s0 += tr.x * m;   a_q0 += tr.x * tr.x * m;
        a_s1 += tr.y * m;   a_q1 += tr.y * tr.y * m;
        a_s2 += tr.z * m;   a_q2 += tr.z * tr.z * m;
        a_s3 += tr.w * m;   a_q3 += tr.w * tr.w * m;
        a_sc += c * m;      a_qc += c * c * m;
    }
    float r_sm = blockSum(a_sm, s_tot);
    float r_sd = blockSum(a_sd, s_tot);
    float r_s0 = blockSum(a_s0, s_tot);
    float r_s1 = blockSum(a_s1, s_tot);
    float r_s2 = blockSum(a_s2, s_tot);
    float r_s3 = blockSum(a_s3, s_tot);
    float r_q0 = blockSum(a_q0, s_tot);
    float r_q1 = blockSum(a_q1, s_tot);
    float r_q2 = blockSum(a_q2, s_tot);
    float r_q3 = blockSum(a_q3, s_tot);
    float r_sc = blockSum(a_sc, s_tot);
    float r_qc = blockSum(a_qc, s_tot);

    if (tid == 0) {
        float st0 = stats[(size_t)b*5+0], st1 = stats[(size_t)b*5+1],
              st2 = stats[(size_t)b*5+2], st4 = stats[(size_t)b*5+4];
        float visible   = fmaxf(r_sm, 1.f);
        float src_total = fmaxf(r_sd, 1.f);
        float src_mean  = src_total / visible;
        float rms       = fmaxf(st2, 1.f);
        float rate_scale = fminf(fmaxf(rms / fmaxf(src_mean, 1.f), 0.55f), 1.95f);
        float speech_budget = src_total * rate_scale;
        float pause_ratio    = fminf(fmaxf(st0, 0.f), 0.49f);
        float boundary_ratio = fminf(fmaxf(st4, 0.f), 1.f);
        float mean_pause     = fmaxf(st1, 0.f);
        float pfr = speech_budget * pause_ratio / fmaxf(1.f - pause_ratio, 0.2f);
        float pfe = visible * boundary_ratio * mean_pause;
        float pause_budget = fminf(fmaxf(0.35f*pfr + 0.65f*pfe, 0.f), speech_budget * 0.8f);

        float inv = 1.f / visible;
        float m0 = r_s0*inv, m1 = r_s1*inv, m2 = r_s2*inv, m3 = r_s3*inv, mc_ = r_sc*inv;
        float v0 = r_q0*inv - m0*m0, v1 = r_q1*inv - m1*m1,
              v2 = r_q2*inv - m2*m2, v3 = r_q3*inv - m3*m3, vc = r_qc*inv - mc_*mc_;
        sc[0] = visible;
        sc[1] = m0; sc[2] = m1; sc[3] = m2; sc[4] = m3;
        sc[5] = rsqrtf(fmaxf(v0, 1e-6f));
        sc[6] = rsqrtf(fmaxf(v1, 1e-6f));
        sc[7] = rsqrtf(fmaxf(v2, 1e-6f));
        sc[8] = rsqrtf(fmaxf(v3, 1e-6f));
        sc[9] = mc_;
        sc[10] = rsqrtf(fmaxf(vc, 1e-6f));
        sc[11] = speech_budget;
        sc[12] = pause_budget;

        // confidence
        float bt = r_s2 * inv;
        float lr_mean = m1;
        float lr_var  = r_q1*inv - lr_mean*lr_mean;
        float conf = 0.2f + 0.3f*fminf(fmaxf(st0,0.f),1.f)
                          + 0.25f*fminf(fmaxf(st4,0.f),1.f)
                          + 0.2f*bt + 0.1f*expf(-fmaxf(lr_var, 0.f)) + 0.05f;
        out_sb[b] = speech_budget;
        out_pb[b] = pause_budget;
        out_cf[b] = fminf(fmaxf(conf, 0.05f), 1.f);
    }
    __syncthreads();

    const float M0 = sc[1], M1 = sc[2], M2 = sc[3], M3 = sc[4];
    const float I0 = sc[5], I1 = sc[6], I2 = sc[7], I3 = sc[8];
    const float MC = sc[9], IC = sc[10];

    // ------------------------------------------------------------------
    // Pass 2: masked cosine accumulators + raw speech scores -> LDS.
    // ------------------------------------------------------------------
    float a_dot = 0.f, a_nx = 0.f, a_ny = 0.f;
    for (int j = 0; j < E; j++) {
        int t = base + j;
        float m = lds_mask[t];
        float d = lds_dur[t];
        float4 tr = ((const float4*)trR)[t];
        float c = cueR[t];
        float z1 = (tr.y - M1) * I1 * m;
        float z3 = (tr.w - M3) * I3 * m;
        float rb = (tr.z - M2) * I2 * m;
        float sp = fminf(fmaxf((c - MC) * IC * m, -1.5f), 1.5f);
        float pf = (t == T - 1) ? 1.f : cueR[t + 1];
        float raw = expf(log1pf(fmaxf(d, 0.f)) + 0.45f*z1 + 0.3f*z3 + 0.2f*pf) * m;
        s_speech[t] = raw;
        a_dot += sp * rb; a_nx += sp * sp; a_ny += rb * rb;
    }
    float r_dot = blockSum(a_dot, s_tot);
    float r_nx  = blockSum(a_nx,  s_tot);
    float r_ny  = blockSum(a_ny,  s_tot);
    if (tid == 0) {
        float xn = sqrtf(fmaxf(r_nx, 1e-6f));
        float yn = sqrtf(fmaxf(r_ny, 1e-6f));
        float cosv = r_dot / fmaxf(xn * yn, 1e-6f);
        float agree = fminf(fmaxf(cosv, -1.f), 1.f);
        float gate = sigmoidf((agree - 0.15f) * 4.0f);
        gate = 0.05f + (0.55f - 0.05f) * gate;
        sc[13] = 0.35f * gate;           // prior coefficient
    }
    __syncthreads();

    // ------------------------------------------------------------------
    // Pass 3: smooth3 sum + pause scores -> LDS.
    // ------------------------------------------------------------------
    const float CP = sc[13];
    float a_ssum = 0.f;
    for (int j = 0; j < E; j++) {
        int t = base + j;
        float m = lds_mask[t];
        float left  = (t > 0)     ? s_speech[t - 1] : 0.f;
        float right = (t < T - 1) ? s_speech[t + 1] : 0.f;
        float sm3 = (left + s_speech[t] + right) * (1.f / 3.f);
        a_ssum += sm3 * m;
        float4 tr = ((const float4*)trR)[t];
        float c = cueR[t];
        float z0 = (tr.x - M0) * I0 * m;
        float rb = (tr.z - M2) * I2 * m;
        float sp = fminf(fmaxf((c - MC) * IC * m, -1.5f), 1.5f);
        s_pause[t] = expf(1.1f*z0 + 1.5f*rb + CP*sp) * m;
    }
    float r_ssum = blockSum(a_ssum, s_tot);
    if (tid == 0) {
        sc[14] = 1.f / fmaxf(r_ssum, 1e-6f);   // inv denom (speech mnorm)
        sc[17] = r_ssum;
        float ratio = ws[0];
        int vis_i = (int)sc[0]; if (vis_i < 1) vis_i = 1;
        int k = (int)rintf(sc[0] * ratio);
        if (k < 1) k = 1;
        if (k > vis_i) k = vis_i;
        su[0] = 0u;             // threshold bit prefix
        su[1] = (unsigned)k;    // remaining k
    }
    __syncthreads();

    // ------------------------------------------------------------------
    // Variable-k top-k: 4-round radix select on positive-float bit order.
    // ------------------------------------------------------------------
    for (int r = 3; r >= 0; r--) {
        int shift = r * 8;
        hist[tid] = 0u;
        __syncthreads();
        unsigned pref = su[0];
        unsigned pmask = (r == 3) ? 0u : (0xFFFFFFFFu << ((r + 1) * 8));
        for (int j = 0; j < E; j++) {
            unsigned u = __float_as_uint(s_pause[base + j]);
            if ((u & pmask) == pref) atomicAdd(&hist[(u >> shift) & 255u], 1u);
        }
        __syncthreads();
        if (tid == 0) {
            unsigned rem = su[1];
            unsigned c = 0; int selb = 0;
            for (int bb = 255; bb >= 0; bb--) {
                unsigned h = hist[bb];
                if (c + h >= rem) { selb = bb; break; }
                c += h;
            }
            su[0] = pref | ((unsigned)selb << shift);
            su[1] = rem - c;    // still needed within selected bucket
        }
        __syncthreads();
    }
    const unsigned thr = su[0];
    const int keep_eq = (int)su[1];

    // Stable tie-break (index order among equals) + kept sum.
    int eqcnt = 0;
    for (int j = 0; j < E; j++)
        eqcnt += (__float_as_uint(s_pause[base + j]) == thr) ? 1 : 0;
    s_tot[tid] = (float)eqcnt;
    __syncthreads();
    if (tid == 0) {
        float run = 0.f;
        for (int i = 0; i < NT; i++) { float t = s_tot[i]; s_tot[i] = run; run += t; }
    }
    __syncthreads();
    int rank = (int)s_tot[tid];
    __syncthreads();
    float a_kept = 0.f;
    for (int j = 0; j < E; j++) {
        int t = base + j;
        float v = s_pause[t];
        unsigned u = __float_as_uint(v);
        bool keep;
        if (u > thr) keep = true;
        else if (u == thr) { keep = (rank < keep_eq); rank++; }
        else keep = false;
        float kv = keep ? v : 0.f;
        s_pause[t] = kv;
        a_kept += kv;
    }
    float r_kept = blockSum(a_kept, s_tot);
    if (tid == 0) {
        float inv_dp = 1.f / fmaxf(r_kept, 1e-6f);
        sc[15] = inv_dp;
        float S_spe = sc[11] * (sc[17] * sc[14]);
        float S_pau = sc[12] * (r_kept * inv_dp);
        sc[16] = 1.f / fmaxf(S_spe + S_pau, 1e-6f);
    }
    __syncthreads();

    // ------------------------------------------------------------------
    // Pass 5a: exec values, allocation, trace_ctx; scan partial totals.
    // ------------------------------------------------------------------
    const float SB = sc[11], PB = sc[12], IDS = sc[14], IDP = sc[15], IAL = sc[16];
    float tot = 0.f;
    for (int j = 0; j < E; j++) {
        int t = base + j;
        float m = lds_mask[t];
        float d = lds_dur[t];
        float left  = (t > 0)     ? s_speech[t - 1] : 0.f;
        float right = (t < T - 1) ? s_speech[t + 1] : 0.f;
        float sm3 = (left + s_speech[t] + right) * (1.f / 3.f);
        float se = sm3 * m * IDS * SB;
        float pe = s_pause[t] * IDP * PB;
        out_se[rowf + t] = se;
        out_pe[rowf + t] = pe;
        out_al[rowf + t] = (se + pe) * IAL;
        float4 tr = ((const float4*)trR)[t];
        float4 tc; tc.x = tr.x*m; tc.y = tr.y*m; tc.z = tr.z*m; tc.w = tr.w*m;
        ((float4*)out_tc)[rowf + t] = tc;
        tot += (se + pe - d) * m;
    }
    s_tot[tid] = tot;
    __syncthreads();
    if (tid == 0) {
        float run = 0.f;
        for (int i = 0; i < NT; i++) { float t = s_tot[i]; s_tot[i] = run; run += t; }
    }
    __syncthreads();

    // Pass 5b: finalize prefix_clock / prefix_backlog.
    float run = s_tot[tid];
    for (int j = 0; j < E; j++) {
        int t = base + j;
        float m = lds_mask[t];
        float d = lds_dur[t];
        float left  = (t > 0)     ? s_speech[t - 1] : 0.f;
        float right = (t < T - 1) ? s_speech[t + 1] : 0.f;
        float sm3 = (left + s_speech[t] + right) * (1.f / 3.f);
        float se = sm3 * m * IDS * SB;
        float pe = s_pause[t] * IDP * PB;
        run += (se + pe - d) * m;
        float pc = run * m;
        out_pc[rowf + t] = pc;
        out_bk[rowf + t] = fmaxf(pc, 0.f) * m;
    }
}

// ---------------------------------------------------------------------------
extern "C" void kernel_launch(void* const* d_in, const int* in_sizes, int n_in,
                              void* d_out, int out_size, void* d_ws, size_t ws_size,
                              hipStream_t stream) {
    const float* dur   = (const float*)d_in[0];
    const float* stats = (const float*)d_in[1];
    const float* trace = (const float*)d_in[2];
    const float* maskp = (const float*)d_in[3];
    const float* cuep  = (const float*)d_in[4];
    int B = in_sizes[1] / 5;
    int T = in_sizes[0] / B;
    float* ws = (float*)d_ws;
    ratio_kernel<<<1, 256, 0, stream>>>(stats, ws, B);
    teacher_kernel<<<B, NT, 0, stream>>>(dur, stats, trace, maskp, cuep,
                                         (float*)d_out, ws, B, T);
}